// DCT2D_3599182594809
// MI455X (gfx1250) — compile-verified
//
#include <hip/hip_runtime.h>
#include <hip/hip_bf16.h>

typedef __attribute__((ext_vector_type(2))) float v2f;
typedef __attribute__((ext_vector_type(8))) float v8f;

#define IMG_W        512
#define TILES_X      32            // 512/16
#define N_TILES      98304         // 32*3*(512/16)*(512/16)
#define TILES_PER_WAVE 4
#define WAVES_PER_BLOCK 8
#define N_BLOCKS     (N_TILES / (TILES_PER_WAVE * WAVES_PER_BLOCK))  // 3072, exact
#define LDS_STRIDE   18            // padded row stride (floats); even -> 8B-aligned b64 reloads

// Dbig = diag(D, D): block-diagonal 16x16 built from the 8x8 DCT basis.
__device__ __forceinline__ float dbig(const float* __restrict__ b, int i, int j) {
    return ((i >> 3) == (j >> 3)) ? b[((i & 7) << 3) + (j & 7)] : 0.0f;
}

__global__ __launch_bounds__(256) void DCT2D_3599182594809_kernel(
    const float* __restrict__ x,
    const float* __restrict__ basis,
    float* __restrict__ out)
{
    __shared__ float lds[WAVES_PER_BLOCK * 16 * LDS_STRIDE];

    const int tid  = threadIdx.x;
    const int wave = tid >> 5;
    const int lane = tid & 31;
    const int m    = lane & 15;      // row (A) / column (B,C,D) index per lane
    const int hi   = lane >> 4;      // lane half
    const int d    = hi << 1;        // K / column offset for this half

    float* tscr = &lds[wave * 16 * LDS_STRIDE];

    // Basis operand registers. By symmetry of the layouts, the same per-lane
    // values serve as A of (Dbig @ X) and as B of (T @ Dbig^T).
    v2f aD[4];
#pragma unroll
    for (int k = 0; k < 4; ++k) {
        const int c0 = 4 * k + d;
        aD[k].x = dbig(basis, m, c0);
        aD[k].y = dbig(basis, m, c0 + 1);
    }

    const int gw = blockIdx.x * WAVES_PER_BLOCK + wave;
    const int t0 = gw * TILES_PER_WAVE;

    // Grid exactly covers N_TILES (3072 blocks * 8 waves * 4 tiles): no bounds
    // check needed, so EXEC is structurally all-1s for every WMMA.
#pragma unroll
    for (int tt = 0; tt < TILES_PER_WAVE; ++tt) {
        const int t = t0 + tt;

        const int plane = t >> 10;               // / (32*32 tiles per 512x512 plane)
        const int ty    = (t >> 5) & 31;
        const int tx    = t & 31;
        // base points at (tile row 0, column m) of this tile
        const size_t base = (size_t)plane * (IMG_W * IMG_W)
                          + (size_t)(ty * 16) * IMG_W + (size_t)(tx * 16) + m;

        // ---- Load X directly in WMMA B-operand layout (dense 64B row
        //      segments), non-temporal: each byte is read exactly once.
        v2f xb[4];
#pragma unroll
        for (int k = 0; k < 4; ++k) {
            const int r0 = 4 * k + d;
            xb[k].x = __builtin_nontemporal_load(&x[base + (size_t)r0 * IMG_W]);
            xb[k].y = __builtin_nontemporal_load(&x[base + (size_t)(r0 + 1) * IMG_W]);
        }

        // ---- GEMM1: T = Dbig @ X  (4 chained f32 WMMAs, K=4 each)
        v8f acc = {};
#pragma unroll
        for (int k = 0; k < 4; ++k) {
            acc = __builtin_amdgcn_wmma_f32_16x16x4_f32(
                false, aD[k], false, xb[k], (short)0, acc, false, false);
        }

        // ---- Bounce T through LDS: C/D layout -> A layout (wave-private, no barrier)
#pragma unroll
        for (int i = 0; i < 8; ++i)
            tscr[(i + 8 * hi) * LDS_STRIDE + m] = acc[i];

        asm volatile("s_wait_dscnt 0" ::: "memory");

        v2f ta[4];
#pragma unroll
        for (int k = 0; k < 4; ++k)
            ta[k] = *(const v2f*)&tscr[m * LDS_STRIDE + 4 * k + d]; // 8B-aligned ds_load_b64

        // ---- GEMM2: Y = T @ Dbig^T (same basis registers serve as B operand)
        v8f y = {};
#pragma unroll
        for (int k = 0; k < 4; ++k) {
            y = __builtin_amdgcn_wmma_f32_16x16x4_f32(
                false, ta[k], false, aD[k], (short)0, y, false, false);
        }

        // ---- Store Y from C/D layout (dense 64B row segments), non-temporal:
        //      each byte is written exactly once, keep it out of L2 residency.
#pragma unroll
        for (int i = 0; i < 8; ++i)
            __builtin_nontemporal_store(y[i], &out[base + (size_t)(i + 8 * hi) * IMG_W]);
    }
}

extern "C" void kernel_launch(void* const* d_in, const int* in_sizes, int n_in,
                              void* d_out, int out_size, void* d_ws, size_t ws_size,
                              hipStream_t stream) {
    const float* x     = (const float*)d_in[0];
    const float* basis = (const float*)d_in[1];
    float* out         = (float*)d_out;
    (void)in_sizes; (void)n_in; (void)out_size; (void)d_ws; (void)ws_size;

    DCT2D_3599182594809_kernel<<<N_BLOCKS, 256, 0, stream>>>(x, basis, out);
}